// Expectation_27204322853261
// MI455X (gfx1250) — compile-verified
//
#include <hip/hip_runtime.h>

typedef __attribute__((ext_vector_type(2))) float v2f;
typedef __attribute__((ext_vector_type(8))) float v8f;

#define AS1 __attribute__((address_space(1)))
#define AS3 __attribute__((address_space(3)))

#define B_  32
#define N_  1024
#define D_  32
#define DD  (D_ * D_)   // 1024 floats per op matrix

#if defined(__has_builtin)
#if __has_builtin(__builtin_amdgcn_global_load_async_to_lds_b32) && \
    __has_builtin(__builtin_amdgcn_s_wait_asynccnt)
#define USE_ASYNC_STAGE 1
#else
#define USE_ASYNC_STAGE 0
#endif
#else
#define USE_ASYNC_STAGE 0
#endif

// WGP-scope prefetch (scope field 0 = pull into ALL cache levels incl. WGP$).
// __builtin_prefetch only reaches SE/SYS scope, which stops at GL2, so use
// inline CDNA5 asm. Addresses are guaranteed in-bounds by the caller's guard,
// which is safe for the non-speculative WGP-scope path.
__device__ __forceinline__ void prefetch_wgp(const void* ptr) {
    asm volatile("global_prefetch_b8 %0, off" :: "v"(ptr));
}

// out[b,n] = sum_k opflat_re[k]*vre[k] - opflat_im[k]*vim[k],
// vre = rho_re^T flat, vim = rho_im^T flat (subtraction done at combine).
//
// Block = 256 threads (8 waves). Waves 0-3 run the RE pass for 16-n tiles 0-3,
// waves 4-7 run the IM pass for the same tiles (4096 waves total -> deep
// memory pipeline, half-length WMMA chains). Each wave computes 16 n via
// V_WMMA_F32_16X16X4_F32 with A = per-lane contiguous float2 of its op row
// (ISA 16x4 f32 A layout) and B = v[k0..k0+3] broadcast from LDS.
__global__ __launch_bounds__(256)
void expectation_wmma(const float* __restrict__ ops_re,
                      const float* __restrict__ ops_im,
                      const float* __restrict__ rhos_re,
                      const float* __restrict__ rhos_im,
                      const int*   __restrict__ prefactor,
                      float* __restrict__ out)
{
    __shared__ float vsm[2 * DD];     // [0..1023] = rho_re^T, [1024..2047] = rho_im^T
    __shared__ float part[8][16];     // per-wave partial results (16 n each)

    const int tid  = threadIdx.x;
    const int wg   = blockIdx.x;      // 0..511
    const int b    = wg >> 4;         // 16 workgroups per batch b
    const int nblk = wg & 15;         // 64-n slab within batch
    const int wave = tid >> 5;
    const int lane = tid & 31;
    const int tile = wave & 3;        // which 16-n tile in the slab
    const int pass = wave >> 2;       // 0 = re, 1 = im
    const int n0   = nblk * 64 + tile * 16;

    // Stage v = [rho_re^T || rho_im^T] into LDS. The transpose is a per-lane
    // gather -> per-lane LDS scatter: exactly GLOBAL_LOAD_ASYNC_TO_LDS_B32.
    {
        const float* rr = rhos_re + (size_t)b * DD;
        const float* ri = rhos_im + (size_t)b * DD;
#if USE_ASYNC_STAGE
        for (int t = tid; t < DD; t += 256) {
            const int i = t >> 5, j = t & 31;      // t = i*32 + j (op flat index)
            __builtin_amdgcn_global_load_async_to_lds_b32(
                (AS1 int*)(rr + j * D_ + i), (AS3 int*)&vsm[t], 0, 0);
            __builtin_amdgcn_global_load_async_to_lds_b32(
                (AS1 int*)(ri + j * D_ + i), (AS3 int*)&vsm[DD + t], 0, 0);
        }
        __builtin_amdgcn_s_wait_asynccnt(0);
#else
        for (int t = tid; t < DD; t += 256) {
            const int i = t >> 5, j = t & 31;
            vsm[t]      = rr[j * D_ + i];
            vsm[DD + t] = ri[j * D_ + i];
        }
#endif
    }
    __syncthreads();

    // A-fragment addressing per ISA 16x4 f32 layout:
    //   lane L: M = L & 15 ; VGPR0/1 hold K = {0,1} (lanes 0-15) or {2,3} (16-31)
    const int row = lane & 15;
    const int ka  = (lane >> 4) * 2;
    const float* p = (pass ? ops_im : ops_re)
                     + ((size_t)(b * N_ + n0 + row)) * DD + (size_t)ka;
    const int voff = pass * DD + ka;

    v8f c0 = {}, c1 = {};             // two chains to break WMMA RAW dependency

    #pragma unroll 1
    for (int k0 = 0; k0 < DD; k0 += 64) {
        if (k0 + 256 < DD) {
            // Cover BOTH cachelines of this block's 256B address span, 1KB
            // ahead, at WGP scope so demand loads hit WGP$.
            prefetch_wgp(p + k0 + 256);
            prefetch_wgp(p + k0 + 288);
        }
        // Batch all 16 A fragments first: 16 global_load_b64 in flight (clause)
        v2f a[16];
        #pragma unroll
        for (int i2 = 0; i2 < 16; ++i2)
            a[i2] = *(const v2f*)(p + k0 + i2 * 4);
        #pragma unroll
        for (int i2 = 0; i2 < 16; i2 += 2) {
            v2f b0 = *(const v2f*)(&vsm[voff + k0 + i2 * 4]);
            v2f b1 = *(const v2f*)(&vsm[voff + k0 + i2 * 4 + 4]);
            c0 = __builtin_amdgcn_wmma_f32_16x16x4_f32(
                    false, a[i2],     false, b0, (short)0, c0, false, false);
            c1 = __builtin_amdgcn_wmma_f32_16x16x4_f32(
                    false, a[i2 + 1], false, b1, (short)0, c1, false, false);
        }
    }

    const v8f c = c0 + c1;

    // C layout: VGPR r, lanes 0-15 -> M=r ; lanes 16-31 -> M=8+r (columns equal).
    if ((lane & 15) == 0) {
        float* dst = &part[wave][(lane >> 4) * 8];
        #pragma unroll
        for (int r = 0; r < 8; ++r) dst[r] = c[r];
    }
    __syncthreads();

    // Combine: out = pf * (re_part - im_part). 64 outputs per workgroup.
    if (tid < 64) {
        const int t16 = tid >> 4;     // tile
        const int m   = tid & 15;     // row within tile
        const float pf = (float)prefactor[0];
        out[(size_t)b * N_ + (size_t)(nblk * 64 + t16 * 16 + m)] =
            pf * (part[t16][m] - part[t16 + 4][m]);
    }
}

extern "C" void kernel_launch(void* const* d_in, const int* in_sizes, int n_in,
                              void* d_out, int out_size, void* d_ws, size_t ws_size,
                              hipStream_t stream) {
    const float* ops_re    = (const float*)d_in[0];
    const float* ops_im    = (const float*)d_in[1];
    const float* rhos_re   = (const float*)d_in[2];
    const float* rhos_im   = (const float*)d_in[3];
    const int*   prefactor = (const int*)d_in[4];
    float* out = (float*)d_out;

    dim3 grid(B_ * (N_ / 64));   // 512 workgroups
    dim3 block(256);             // 8 waves: 4 re-waves + 4 im-waves
    hipLaunchKernelGGL(expectation_wmma, grid, block, 0, stream,
                       ops_re, ops_im, rhos_re, rhos_im, prefactor, out);
}